// ImageWarpingLayer_9749575762160
// MI455X (gfx1250) — compile-verified
//
#include <hip/hip_runtime.h>

// ---------------------------------------------------------------------------
// Depth-of-field splat + integral-image pipeline for MI455X (gfx1250).
//
//  Phase 1: corner scatter with global f32 atomics (deltas).
//           acc deltas are built directly inside d_out (2*B*C*H*W floats),
//           count deltas in d_ws (2*B*H*W floats).
//  Phase 2: 2-D inclusive prefix sums done as blocked triangular WMMAs:
//             row cumsum :  D = T * U + Carry   (U upper-tri ones, data = A)
//             col cumsum :  D = L * T + Carry   (L lower-tri ones, data = B)
//           using V_WMMA_F32_16X16X4_F32, K=16 as 4 chained K=4 WMMAs.
//           One wave32 per 16-wide strip carries the serial dependency.
//  Phase 3: out = clip(acc / max(cnt,1), 0, 1), in place on d_out.
// ---------------------------------------------------------------------------

#define BATCH 8
#define CHANS 3
#define HH 1024
#define WW 1024

typedef float v2f __attribute__((ext_vector_type(2)));
typedef float v8f __attribute__((ext_vector_type(8)));

static constexpr size_t PLANE = (size_t)HH * WW; // 1 Mi elements

// ---------------------------------------------------------------------------
// Zero-fill (d_out / d_ws are poisoned by the harness)
// ---------------------------------------------------------------------------
__global__ void __launch_bounds__(256) zero_f4(float4* __restrict__ p, size_t n4) {
  size_t i = (size_t)blockIdx.x * blockDim.x + threadIdx.x;
  if (i < n4) p[i] = make_float4(0.f, 0.f, 0.f, 0.f);
}

// ---------------------------------------------------------------------------
// Phase 1: 4-corner delta scatter for left and right shifted boxes.
// ---------------------------------------------------------------------------
__global__ void __launch_bounds__(256) splat_kernel(const float* __restrict__ image,
                                                    const float* __restrict__ depth,
                                                    float* __restrict__ acc,   // d_out
                                                    float* __restrict__ cnt) { // d_ws
  int gid = blockIdx.x * 256 + threadIdx.x;
  if (gid >= BATCH * (int)PLANE) return;
  int x = gid & (WW - 1);
  int y = (gid >> 10) & (HH - 1);
  int b = gid >> 20;

  int R = (int)floorf(depth[gid]);
  size_t ib = (size_t)b * CHANS * PLANE + (size_t)y * WW + x;
  float p0 = image[ib];
  float p1 = image[ib + PLANE];
  float p2 = image[ib + 2 * PLANE];

  int y0c = max(y - R, 0);
  int yp  = y + R + 1;
  bool my = (yp < HH);
  int ypc = min(yp, HH - 1);

#pragma unroll
  for (int v = 0; v < 2; ++v) {
    int xlo = v ? x : x - 2 * R;
    int xhi = v ? x + 2 * R : x;
    int x0c = max(xlo, 0);
    int xp  = xhi + 1;
    bool mx = (xp < WW);
    int xpc = min(xp, WW - 1);

    float* cb = cnt + (size_t)(v * BATCH + b) * PLANE;
    float* ab = acc + (size_t)(v * BATCH + b) * CHANS * PLANE;

    auto corner = [&](int yy, int xx, float w) {
      size_t o = (size_t)yy * WW + xx;
      atomicAdd(cb + o, w);                 // global_atomic_add_f32
      atomicAdd(ab + o,             w * p0);
      atomicAdd(ab + o + PLANE,     w * p1);
      atomicAdd(ab + o + 2 * PLANE, w * p2);
    };

    corner(y0c, x0c, 1.0f);
    if (mx) corner(y0c, xpc, -1.0f);
    if (my) corner(ypc, x0c, -1.0f);
    if (my && mx) corner(ypc, xpc, 1.0f);
  }
}

// ---------------------------------------------------------------------------
// Phase 2a: row cumsum (along w) via WMMA:  D = T * U + Carry.
// One wave32 per (plane, 16-row strip); walks 64 tiles serially.
//
// f32 16x4 A layout (ISA 7.12.2): lane l holds row M=l%16; VGPR v holds
// K = v + 2*(l/16). B 4x16 mirrors: VGPR v, lane l -> (K = v+2*(l/16), N=l%16).
// C/D: VGPR r, lane l -> (M = r + 8*(l/16), N = l%16).
// ---------------------------------------------------------------------------
__global__ void __launch_bounds__(32) rowscan_wmma(float* __restrict__ data) {
  const int strip = blockIdx.x & 63;        // HH/16 strips per plane
  const int plane = blockIdx.x >> 6;
  float* base = data + (size_t)plane * PLANE + (size_t)strip * 16 * WW;

  const int lane = threadIdx.x;
  const int m    = lane & 15;               // row for A, column for D
  const int half = lane >> 4;

  // Constant B operand: U[K][N] = (K <= N);  K = 4k + v + 2*half, N = m.
  v2f bU[4];
#pragma unroll
  for (int k = 0; k < 4; ++k) {
    bU[k].x = ((4 * k + 0 + 2 * half) <= m) ? 1.0f : 0.0f;
    bU[k].y = ((4 * k + 1 + 2 * half) <= m) ? 1.0f : 0.0f;
  }

  // Carry matrix: every column equals running row-sums -> per lane, 8 floats
  // for rows M = r + 8*half.
  v8f carry;
#pragma unroll
  for (int r = 0; r < 8; ++r) carry[r] = 0.0f;

  for (int j = 0; j < WW / 16; ++j) {
    float* tile = base + j * 16;
    if (j + 1 < WW / 16)
      __builtin_prefetch(tile + 16 + (size_t)m * WW, 0, 3); // global_prefetch_b8

    v8f d = carry;
#pragma unroll
    for (int k = 0; k < 4; ++k) {
      // A chunk: lane holds (row m, cols 4k + 2*half + {0,1}) -> 8B load.
      v2f a = *reinterpret_cast<const v2f*>(tile + (size_t)m * WW + 4 * k + 2 * half);
      d = __builtin_amdgcn_wmma_f32_16x16x4_f32(false, a, false, bU[k],
                                                (short)0, d, false, false);
    }

    // Store D in place: lane writes (M = r + 8*half, N = m).
#pragma unroll
    for (int r = 0; r < 8; ++r)
      tile[(size_t)(r + 8 * half) * WW + m] = d[r];

    // Carry update: new carry[M] = D[M][15], held in lanes 15 / 31.
#pragma unroll
    for (int r = 0; r < 8; ++r)
      carry[r] = __shfl(d[r], 15 + 16 * half, 32);
  }
}

// ---------------------------------------------------------------------------
// Phase 2b: column cumsum (along h) via WMMA:  D = L * T + Carry.
// One wave32 per (plane, 16-column strip); walks 64 tiles down serially.
// ---------------------------------------------------------------------------
__global__ void __launch_bounds__(32) colscan_wmma(float* __restrict__ data) {
  const int strip = blockIdx.x & 63;        // WW/16 strips per plane
  const int plane = blockIdx.x >> 6;
  float* base = data + (size_t)plane * PLANE + strip * 16;

  const int lane = threadIdx.x;
  const int n    = lane & 15;               // column index
  const int half = lane >> 4;

  // Constant A operand: L[M][K] = (K <= M); M = n (l%16), K = 4k + v + 2*half.
  v2f aL[4];
#pragma unroll
  for (int k = 0; k < 4; ++k) {
    aL[k].x = ((4 * k + 0 + 2 * half) <= n) ? 1.0f : 0.0f;
    aL[k].y = ((4 * k + 1 + 2 * half) <= n) ? 1.0f : 0.0f;
  }

  float carry = 0.0f;                       // per-column running sum

  for (int i = 0; i < HH / 16; ++i) {
    float* tile = base + (size_t)i * 16 * WW;
    if (i + 1 < HH / 16)
      __builtin_prefetch(tile + (size_t)16 * WW + n, 0, 3);

    v8f d;
#pragma unroll
    for (int r = 0; r < 8; ++r) d[r] = carry;   // rows all equal carry[n]

#pragma unroll
    for (int k = 0; k < 4; ++k) {
      // B chunk: lane holds (rows 4k + 2*half + {0,1}, col n) -> coalesced b32s.
      v2f bT;
      bT.x = tile[(size_t)(4 * k + 0 + 2 * half) * WW + n];
      bT.y = tile[(size_t)(4 * k + 1 + 2 * half) * WW + n];
      d = __builtin_amdgcn_wmma_f32_16x16x4_f32(false, aL[k], false, bT,
                                                (short)0, d, false, false);
    }

#pragma unroll
    for (int r = 0; r < 8; ++r)
      tile[(size_t)(r + 8 * half) * WW + n] = d[r];

    // new carry[n] = D[15][n] : M=15 -> VGPR 7, lanes 16..31.
    carry = __shfl(d[7], n + 16, 32);
  }
}

// ---------------------------------------------------------------------------
// Phase 3: out = clip(acc / max(cnt, 1), 0, 1), float4-vectorized in place.
// out layout: [view][b][c][h][w]; cnt layout: [view][b][h][w].
// ---------------------------------------------------------------------------
__global__ void __launch_bounds__(256) finalize_kernel(float4* __restrict__ out,
                                                       const float4* __restrict__ cnt,
                                                       size_t n4) {
  size_t i = (size_t)blockIdx.x * blockDim.x + threadIdx.x;
  if (i >= n4) return;
  constexpr size_t P4 = PLANE / 4;
  size_t p4 = i % P4;
  size_t pc = i / P4;         // (view*BATCH + b)*CHANS + c
  size_t vb = pc / CHANS;     // view*BATCH + b

  float4 c = cnt[vb * P4 + p4];
  float4 a = out[i];
  a.x = fminf(fmaxf(a.x / fmaxf(c.x, 1.0f), 0.0f), 1.0f);
  a.y = fminf(fmaxf(a.y / fmaxf(c.y, 1.0f), 0.0f), 1.0f);
  a.z = fminf(fmaxf(a.z / fmaxf(c.z, 1.0f), 0.0f), 1.0f);
  a.w = fminf(fmaxf(a.w / fmaxf(c.w, 1.0f), 0.0f), 1.0f);
  out[i] = a;
}

// ---------------------------------------------------------------------------
extern "C" void kernel_launch(void* const* d_in, const int* in_sizes, int n_in,
                              void* d_out, int out_size, void* d_ws, size_t ws_size,
                              hipStream_t stream) {
  const float* image = (const float*)d_in[0];   // (B,C,H,W) f32
  const float* depth = (const float*)d_in[1];   // (B,H,W)   f32
  float* out = (float*)d_out;                   // left ++ right, (2,B,C,H,W)
  float* cnt = (float*)d_ws;                    // (2,B,H,W) counts (64 MB)

  const size_t outN = (size_t)2 * BATCH * CHANS * PLANE;  // 50,331,648
  const size_t cntN = (size_t)2 * BATCH * PLANE;          // 16,777,216

  zero_f4<<<dim3((unsigned)((outN / 4 + 255) / 256)), 256, 0, stream>>>(
      (float4*)out, outN / 4);
  zero_f4<<<dim3((unsigned)((cntN / 4 + 255) / 256)), 256, 0, stream>>>(
      (float4*)cnt, cntN / 4);

  splat_kernel<<<dim3((unsigned)(BATCH * PLANE / 256)), 256, 0, stream>>>(
      image, depth, out, cnt);

  // Row cumsum: counts (2*B = 16 planes), then accs (2*B*C = 48 planes).
  rowscan_wmma<<<dim3(16 * (HH / 16)), 32, 0, stream>>>(cnt);
  rowscan_wmma<<<dim3(48 * (HH / 16)), 32, 0, stream>>>(out);

  // Column cumsum.
  colscan_wmma<<<dim3(16 * (WW / 16)), 32, 0, stream>>>(cnt);
  colscan_wmma<<<dim3(48 * (WW / 16)), 32, 0, stream>>>(out);

  finalize_kernel<<<dim3((unsigned)((outN / 4 + 255) / 256)), 256, 0, stream>>>(
      (float4*)out, (const float4*)cnt, outN / 4);
}